// Yolo_v3_76828374991005
// MI455X (gfx1250) — compile-verified
//
#include <hip/hip_runtime.h>
#include <math.h>

#define NCLS 20
#define NA 3
#define DIMT 10647
#define BATCH 64
#define NGT 8
#define EPS16 6.103515625e-05f

__device__ __constant__ float c_AW[9] = {0.025f,0.04f,0.08f,0.07f,0.15f,0.14f,0.28f,0.38f,0.9f};
__device__ __constant__ float c_AH[9] = {0.033f,0.07f,0.06f,0.15f,0.11f,0.29f,0.22f,0.48f,0.78f};
__device__ __constant__ int   c_GRID[3]  = {52, 26, 13};
__device__ __constant__ int   c_COFF[3]  = {0, 2704, 3380};

__device__ __forceinline__ float sigm(float x) { return 1.0f / (1.0f + expf(-x)); }
__device__ __forceinline__ float bce(float x, float t) {
    return fmaxf(x, 0.0f) - x * t + log1pf(expf(-fabsf(x)));
}

// ---------------- Kernel 1: per-image partial sums of sigmoid(pconf)^2 ----------------
__global__ void k_conf_partial(const float* __restrict__ pconf, float* __restrict__ part) {
    __shared__ float red[256];
    const int b = blockIdx.x;
    const float* p = pconf + (size_t)b * DIMT;
    float acc = 0.0f;
    for (int i = threadIdx.x; i < DIMT; i += 256) {
        float s = sigm(p[i]);
        acc += s * s;
    }
    red[threadIdx.x] = acc;
    __syncthreads();
    for (int off = 128; off > 0; off >>= 1) {
        if (threadIdx.x < off) red[threadIdx.x] += red[threadIdx.x + off];
        __syncthreads();
    }
    if (threadIdx.x == 0) part[b] = red[0];
}

// ---------------- Kernel 2: one wave per image — match + sparse loss terms ----------------
__global__ void __launch_bounds__(32) k_match(const float* __restrict__ pconf,
                                              const float* __restrict__ pcls,
                                              const float* __restrict__ ptxywh,
                                              const float* __restrict__ gboxes,
                                              const int*   __restrict__ glabels,
                                              float* __restrict__ imgout) {
    const int b = blockIdx.x;
    const int lane = threadIdx.x;

    __shared__ int   s_base[NGT][3];
    __shared__ float s_txy[NGT][3][2];
    __shared__ float s_w[NGT], s_h[NGT], s_wt[NGT];
    __shared__ int   s_ids[NGT], s_ceng[NGT], s_aoff[NGT], s_lab[NGT];
    __shared__ int   t_idx[80];
    __shared__ int   t_conf[80];
    __shared__ int   t_si[80], t_sj[80];
    __shared__ int   s_n;

    if (lane < NGT) {
        const int j = lane;
        const float* gb = gboxes + ((size_t)b * NGT + j) * 4;
        float l = gb[0], t = gb[1], r = gb[2], bo = gb[3];
        float cx = (l + r) * 0.5f, cy = (t + bo) * 0.5f;
        float w = r - l, h = bo - t;
        // anchor IoU argmax (first-max wins: strict >)
        int best = 0; float bestv = -1.0f;
        for (int k = 0; k < 9; k++) {
            float inter = fminf(w, c_AW[k]) * fminf(h, c_AH[k]);
            float iou = inter / (w * h + c_AW[k] * c_AH[k] - inter);
            if (iou > bestv) { bestv = iou; best = k; }
        }
        s_ids[j] = best;
        s_ceng[j] = best / NA;
        s_aoff[j] = best % NA;
        s_lab[j] = glabels[b * NGT + j] - 1;
        s_w[j] = w; s_h[j] = h;
        s_wt[j] = 2.0f - w * h;
        for (int lyr = 0; lyr < 3; lyr++) {
            float gf = (float)c_GRID[lyr];
            int col = (int)(cx * gf);   // trunc toward zero (cx > 0)
            int row = (int)(cy * gf);
            s_base[j][lyr] = (c_COFF[lyr] + row * c_GRID[lyr] + col) * NA;
            s_txy[j][lyr][0] = (cx - (float)col / gf) * gf;
            s_txy[j][lyr][1] = (cy - (float)row / gf) * gf;
        }
    }
    __syncthreads();

    // Faithful sequential scan simulation (k = i*8 + j; neg marks then pos write)
    if (lane == 0) {
        int n = 0;
        for (int k = 0; k < NGT * NGT; k++) {
            int i = k >> 3, j = k & 7;
            for (int lyr = 0; lyr < 3; lyr++) {
                int bb = s_base[i][lyr];
                for (int a = 0; a < 3; a++) {
                    int d = bb + a;
                    int e = -1;
                    for (int q = 0; q < n; q++) if (t_idx[q] == d) { e = q; break; }
                    if (e < 0) { e = n; t_idx[n] = d; n++; }
                    t_conf[e] = -1;
                }
            }
            int p = s_base[i][s_ceng[j]] + s_aoff[j];   // always within neg[i], so present
            for (int q = 0; q < n; q++)
                if (t_idx[q] == p) { t_conf[q] = 1; t_si[q] = i; t_sj[q] = j; break; }
        }
        s_n = n;
    }
    __syncthreads();

    const int n = s_n;
    float a_negcorr = 0.0f, a_pos = 0.0f, a_cls = 0.0f, a_txty = 0.0f, a_twth = 0.0f;
    float a_poscnt = 0.0f, a_nzcnt = 0.0f;

    for (int e = lane; e < n; e += 32) {
        const int d = t_idx[e];
        float s = sigm(pconf[(size_t)b * DIMT + d]);
        a_negcorr += s * s;     // every touched slot has gconf != 0
        a_nzcnt += 1.0f;
        if (t_conf[e] == 1) {
            a_poscnt += 1.0f;
            float dm = s - 1.0f;
            a_pos += dm * dm;
            const int i = t_si[e], j = t_sj[e];
            // class BCE vs one-hot(label[i])
            const float* pc = pcls + ((size_t)b * DIMT + d) * NCLS;
            const int lab = s_lab[i];
            float cs = 0.0f;
            for (int c = 0; c < NCLS; c++)
                cs += bce(pc[c], (c == lab) ? 1.0f : 0.0f);
            a_cls += cs;
            // txty BCE (layer from j, cell data from i) and twth MSE (wh[i] / anc[ids[j]])
            const int lyr = s_ceng[j];
            const float* pt = ptxywh + ((size_t)b * DIMT + d) * 4;
            float bt = bce(pt[0], s_txy[i][lyr][0]) + bce(pt[1], s_txy[i][lyr][1]);
            a_txty += bt * s_wt[i];
            float tw = logf(s_w[i] / c_AW[s_ids[j]]);
            float th = logf(s_h[i] / c_AH[s_ids[j]]);
            float d2 = pt[2] - tw, d3 = pt[3] - th;
            a_twth += (d2 * d2 + d3 * d3) * s_wt[i];
        }
    }

    float vals[7] = {a_negcorr, a_pos, a_poscnt, a_nzcnt, a_cls, a_txty, a_twth};
    for (int q = 0; q < 7; q++) {
        float v = vals[q];
        for (int m = 16; m > 0; m >>= 1) v += __shfl_xor(v, m, 32);
        vals[q] = v;
    }
    if (lane == 0)
        for (int q = 0; q < 7; q++) imgout[b * 8 + q] = vals[q];
}

// ---------------- Kernel 3: finalize (WMMA f32 reduction of the 64 partials) ----------------
typedef __attribute__((ext_vector_type(2))) float v2f;
typedef __attribute__((ext_vector_type(8))) float v8f;

__global__ void __launch_bounds__(32) k_finalize(const float* __restrict__ ws,
                                                 float* __restrict__ out) {
    const int lane = threadIdx.x;
    const float* part = ws;        // [64] pconf sigmoid^2 partials
    const float* img  = ws + 64;   // [64][8] per-image terms

    float S_all = 0.0f;
#if __has_builtin(__builtin_amdgcn_wmma_f32_16x16x4_f32)
    // D = ones(16x4) x B(4x16): each D column = exact f32 sum of 4 B entries.
    // Packing of the 64 partials into B lanes is irrelevant since A is all-ones.
    v2f Av; Av.x = 1.0f; Av.y = 1.0f;
    v2f Bv; Bv.x = part[2 * lane]; Bv.y = part[2 * lane + 1];
    v8f Cv = {};
    v8f Dv = __builtin_amdgcn_wmma_f32_16x16x4_f32(
        /*neg_a=*/false, Av, /*neg_b=*/false, Bv,
        /*c_mod=*/(short)0, Cv, /*reuse_a=*/false, /*reuse_b=*/false);
    float v = Dv[0];                      // lanes 0..15 hold the 16 column sums
    v += __shfl_xor(v, 1, 32);
    v += __shfl_xor(v, 2, 32);
    v += __shfl_xor(v, 4, 32);
    v += __shfl_xor(v, 8, 32);
    S_all = v;                            // valid on lane 0 (sum of lanes 0..15)
#else
    if (lane == 0) { float a = 0.0f; for (int i = 0; i < 64; i++) a += part[i]; S_all = a; }
#endif

    if (lane == 0) {
        float gNegCorr = 0.0f, gPos = 0.0f, gPosCnt = 0.0f, gNz = 0.0f;
        float accCls = 0.0f, accTxty = 0.0f, accTwth = 0.0f;
        for (int bb = 0; bb < BATCH; bb++) {
            const float* r = img + bb * 8;
            gNegCorr += r[0];
            gPos     += r[1];
            gPosCnt  += r[2];
            gNz      += r[3];
            float numsPos = fmaxf(r[2], EPS16);
            accCls  += r[4] / numsPos;
            accTxty += r[5] / numsPos;
            accTwth += r[6] / numsPos;
        }
        const float cntAll = (float)BATCH * (float)DIMT;
        float negSum = S_all - gNegCorr;          // sum sigmoid^2 at gconf==0 slots
        float negCnt = cntAll - gNz;
        float l_conf_pos = gPos / fmaxf(gPosCnt, EPS16) * 30.0f;
        float l_conf_neg = negSum / fmaxf(negCnt, EPS16) * 30.0f;
        float l_cls  = accCls  / (float)BATCH;
        float l_txty = accTxty / (float)BATCH;
        float l_twth = accTwth / (float)BATCH;
        out[0] = l_conf_pos + l_conf_neg + l_cls + l_txty + l_twth;
    }
}

extern "C" void kernel_launch(void* const* d_in, const int* in_sizes, int n_in,
                              void* d_out, int out_size, void* d_ws, size_t ws_size,
                              hipStream_t stream) {
    const float* pconf  = (const float*)d_in[0];
    const float* pcls   = (const float*)d_in[1];
    const float* ptxywh = (const float*)d_in[2];
    const float* gboxes = (const float*)d_in[3];
    const int*   glabels = (const int*)d_in[4];
    float* ws  = (float*)d_ws;    // [0..63] partials, [64..575] per-image terms
    float* out = (float*)d_out;

    k_conf_partial<<<BATCH, 256, 0, stream>>>(pconf, ws);
    k_match<<<BATCH, 32, 0, stream>>>(pconf, pcls, ptxywh, gboxes, glabels, ws + 64);
    k_finalize<<<1, 32, 0, stream>>>(ws, out);
}